// FLAME_37203006718550
// MI455X (gfx1250) — compile-verified
//
#include <hip/hip_runtime.h>
#include <math.h>

// ---------------------------------------------------------------------------
// FLAME forward on MI455X (gfx1250): fused shape+pose GEMM via f32 WMMA,
// fused LBS epilogue, tiny per-batch FK kernel.
// W is stored pre-swizzled in WMMA B-fragment order so every B-frag is one
// 8-byte global_load_b64.
// ---------------------------------------------------------------------------

typedef __attribute__((ext_vector_type(2))) float v2f;
typedef __attribute__((ext_vector_type(8))) float v8f;

#define B_      2048
#define V_      5023
#define N3      15069          // V*3
#define NCOLP   15072          // padded to multiple of 48
#define KF      96             // padded K: 50 expr + 36 pose_feature + 10 zero
#define NT      314            // n-tiles of width 48 (314*48 = 15072)
#define MT      128            // m-tiles of 16 batches
#define PF_OFF  (B_ * V_ * 3)  // 30861312
#define A_OFF   (PF_OFF + B_ * 36)

// ws layout (floats)
#define WS_F    0                      // F[B_][KF]              196608
#define WS_W    (WS_F + B_ * KF)       // Wsw swizzled [96*15072] 1446912
#define WS_JT   (WS_W + KF * NCOLP)    // jt[16]
#define WS_JS   (WS_JT + 16)           // js[5*3*50]

// ---------------------------------------------------------------------------
// Kernel 1: pre-contract J_regressor with v_template and shapedirs slice.
//   jt[j*3+k]          = sum_v Jreg[j,v] * v_template[v,k]
//   js[(j*3+k)*50 + e] = sum_v Jreg[j,v] * shapedirs[v,k,100+e]
// ---------------------------------------------------------------------------
__global__ __launch_bounds__(256) void build_joint_bases(
    const float* __restrict__ Jreg, const float* __restrict__ vtemp,
    const float* __restrict__ shapedirs, float* __restrict__ jt,
    float* __restrict__ js) {
  int o = blockIdx.x * 256 + threadIdx.x;
  if (o < 15) {
    int j = o / 3, k = o % 3;
    float s = 0.f;
    for (int v = 0; v < V_; ++v) s += Jreg[j * V_ + v] * vtemp[v * 3 + k];
    jt[o] = s;
  } else if (o < 765) {
    int idx = o - 15;
    int j = idx / 150, rem = idx % 150, k = rem / 50, e = rem % 50;
    float s = 0.f;
    for (int v = 0; v < V_; ++v)
      s += Jreg[j * V_ + v] * shapedirs[v * 450 + k * 150 + 100 + e];
    js[idx] = s;
  }
}

// ---------------------------------------------------------------------------
// Kernel 2: pack basis matrix into WMMA-B-fragment-swizzled layout.
// Logical W[f][n], f in 0..95:
//   rows 0..49  : shapedirs[v,k,100+f] at n = v*3+k   (expression basis)
//   rows 50..85 : posedirs[f-50, n]                    (pose basis)
//   rows 86..95 and cols 15069..15071: zero padding
// Swizzle: f = 4q + 2*half + r  ->  Wsw[((q*2 + half)*NCOLP + n)*2 + r]
// so a lane's (k,k+1) B-frag pair is a single aligned float2.
// ---------------------------------------------------------------------------
__global__ __launch_bounds__(256) void build_W(
    const float* __restrict__ shapedirs, const float* __restrict__ posedirs,
    float* __restrict__ Wsw) {
  int d = blockIdx.x * 256 + threadIdx.x;
  if (d >= KF * NCOLP) return;
  int r = d & 1;
  int n = (d >> 1) % NCOLP;
  int row = (d >> 1) / NCOLP;       // row = q*2 + half, in 0..47
  int half = row & 1;
  int q = row >> 1;
  int f = 4 * q + 2 * half + r;     // logical basis row
  float val = 0.f;
  if (n < N3) {
    if (f < 50) {
      int v = n / 3, k = n % 3;
      val = shapedirs[v * 450 + k * 150 + 100 + f];
    } else if (f < 86) {
      val = posedirs[(f - 50) * N3 + n];
    }
  }
  Wsw[d] = val;
}

__device__ __forceinline__ void mm4(const float* __restrict__ A,
                                    const float* __restrict__ B,
                                    float* __restrict__ C) {
#pragma unroll
  for (int r = 0; r < 4; ++r)
#pragma unroll
    for (int c = 0; c < 4; ++c) {
      float s = 0.f;
#pragma unroll
      for (int k = 0; k < 4; ++k) s += A[r * 4 + k] * B[k * 4 + c];
      C[r * 4 + c] = s;
    }
}

// ---------------------------------------------------------------------------
// Kernel 3: per-batch Rodrigues + pose_feature + joints + FK chain.
// One thread per batch element. Writes:
//   F[b][0:50]=expr, F[b][50:86]=pose_feature, F[b][86:96]=0
//   pf_out[b][36], A_out[b][5][16] (relative transforms)
// ---------------------------------------------------------------------------
__global__ __launch_bounds__(256) void flame_pose_fk(
    const float* __restrict__ expr, const float* __restrict__ pose,
    const float* __restrict__ jt, const float* __restrict__ js,
    float* __restrict__ F, float* __restrict__ pf_out,
    float* __restrict__ A_out) {
  int b = blockIdx.x * blockDim.x + threadIdx.x;
  if (b >= B_) return;

  float e[50];
#pragma unroll
  for (int i = 0; i < 50; ++i) {
    e[i] = expr[b * 50 + i];
    F[b * KF + i] = e[i];
  }

  // Rodrigues (K^2 = r r^T - (r.r) I identity, exact for any r)
  float R[5][9];
#pragma unroll
  for (int j = 0; j < 5; ++j) {
    float vx = pose[b * 15 + 3 * j + 0];
    float vy = pose[b * 15 + 3 * j + 1];
    float vz = pose[b * 15 + 3 * j + 2];
    float ax = vx + 1e-8f, ay = vy + 1e-8f, az = vz + 1e-8f;
    float ang = sqrtf(ax * ax + ay * ay + az * az);
    float inv = 1.f / ang;
    float rx = vx * inv, ry = vy * inv, rz = vz * inv;
    float s = sinf(ang), c = cosf(ang), o = 1.f - c;
    float rr = rx * rx + ry * ry + rz * rz;
    R[j][0] = 1.f + o * (rx * rx - rr);
    R[j][1] = -s * rz + o * (rx * ry);
    R[j][2] = s * ry + o * (rx * rz);
    R[j][3] = s * rz + o * (rx * ry);
    R[j][4] = 1.f + o * (ry * ry - rr);
    R[j][5] = -s * rx + o * (ry * rz);
    R[j][6] = -s * ry + o * (rx * rz);
    R[j][7] = s * rx + o * (ry * rz);
    R[j][8] = 1.f + o * (rz * rz - rr);
  }

  // pose_feature = (R[1:5] - I).flatten
#pragma unroll
  for (int j = 1; j < 5; ++j)
#pragma unroll
    for (int i = 0; i < 9; ++i) {
      float v = R[j][i] - ((i == 0 || i == 4 || i == 8) ? 1.f : 0.f);
      pf_out[b * 36 + (j - 1) * 9 + i] = v;
      F[b * KF + 50 + (j - 1) * 9 + i] = v;
    }
#pragma unroll
  for (int i = 86; i < 96; ++i) F[b * KF + i] = 0.f;

  // joints[j][k] = jt + js . expr
  float J[5][3];
#pragma unroll
  for (int j = 0; j < 5; ++j)
#pragma unroll
    for (int k = 0; k < 3; ++k) {
      float s = jt[j * 3 + k];
      const float* jsp = js + (j * 3 + k) * 50;
#pragma unroll 10
      for (int q = 0; q < 50; ++q) s += jsp[q] * e[q];
      J[j][k] = s;
    }

  // local 4x4 transforms (rel joints), FK chain: parents = [-1,0,1,1,1]
  float T[5][16], G[5][16];
#pragma unroll
  for (int j = 0; j < 5; ++j) {
    int p = (j == 0) ? -1 : (j == 1 ? 0 : 1);
    float tx = J[j][0], ty = J[j][1], tz = J[j][2];
    if (p >= 0) { tx -= J[p][0]; ty -= J[p][1]; tz -= J[p][2]; }
    T[j][0] = R[j][0]; T[j][1] = R[j][1]; T[j][2] = R[j][2]; T[j][3] = tx;
    T[j][4] = R[j][3]; T[j][5] = R[j][4]; T[j][6] = R[j][5]; T[j][7] = ty;
    T[j][8] = R[j][6]; T[j][9] = R[j][7]; T[j][10] = R[j][8]; T[j][11] = tz;
    T[j][12] = 0.f; T[j][13] = 0.f; T[j][14] = 0.f; T[j][15] = 1.f;
  }
#pragma unroll
  for (int i = 0; i < 16; ++i) G[0][i] = T[0][i];
  mm4(G[0], T[1], G[1]);
  mm4(G[1], T[2], G[2]);
  mm4(G[1], T[3], G[3]);
  mm4(G[1], T[4], G[4]);

  // A = G with translation column minus R_chain * joint (rest-pose removal)
#pragma unroll
  for (int j = 0; j < 5; ++j) {
    float jx = J[j][0], jy = J[j][1], jz = J[j][2];
#pragma unroll
    for (int r = 0; r < 4; ++r) {
      float g0 = G[j][r * 4 + 0], g1 = G[j][r * 4 + 1], g2 = G[j][r * 4 + 2];
      float t3 = G[j][r * 4 + 3] - (g0 * jx + g1 * jy + g2 * jz);
      float* Ao = A_out + b * 80 + j * 16 + r * 4;
      Ao[0] = g0; Ao[1] = g1; Ao[2] = g2; Ao[3] = t3;
    }
  }
}

// ---------------------------------------------------------------------------
// Kernel 4: fused GEMM (v_posed offsets) + LBS epilogue.
// Each wave computes a 16(batch) x 48(=16 verts x xyz) tile of
//   v_posed = F[16x96] x W[96x48]  via 24 k-steps of V_WMMA_F32_16X16X4_F32
// (1 b64 A-frag load + 3 b64 swizzled B-frag loads per k-step),
// stages it to LDS, then blends skinning transforms and writes verts.
// ---------------------------------------------------------------------------
__global__ __launch_bounds__(256) void flame_gemm_lbs(
    const float* __restrict__ F, const float* __restrict__ Wsw,
    const float* __restrict__ A,       // [B_][5][16] (top 3 rows used)
    const float* __restrict__ vtemp,   // [V_][3]
    const float* __restrict__ lbsw,    // [V_][5]
    float* __restrict__ verts) {       // [B_][V_][3]
  // per-wave LDS: 16x48 tile (768) + 16 batches x 5 joints x 12 A-entries (960)
  __shared__ float smem[8 * 1728];
  const int lane = threadIdx.x & 31;
  const int wave = threadIdx.x >> 5;
  const int tile = blockIdx.x * 8 + wave;   // grid sized exactly: MT*NT/8
  const int mt = tile / NT;
  const int nt = tile % NT;
  const int m0 = mt * 16;      // batch base
  const int n0 = nt * 48;      // padded column base
  const int v0 = nt * 16;      // vertex base

  float* tileL = smem + wave * 1728;
  float* aL = tileL + 768;

  // stage A top-3-rows for the 16 batches of this tile into LDS
  for (int i = lane; i < 960; i += 32) {
    int bb = i / 60, rest = i % 60, j = rest / 12, r = rest % 12;
    aL[i] = A[(m0 + bb) * 80 + j * 16 + r];
  }

  const int half = lane >> 4;    // selects K-pair within each k-quad
  const int col = lane & 15;     // M row for A-frag, N col for B-frag

  v8f c0 = {}, c1 = {}, c2 = {};
  const float* Fp = F + (m0 + col) * KF + half * 2;
  // swizzled B base: (row = q*2 + half, n) -> float2 at ((row)*NCOLP + n)*2
  const float* Wp = Wsw + ((size_t)half * NCOLP + n0 + col) * 2;

  for (int k0 = 0; k0 < KF; k0 += 4) {
    v2f a = *(const v2f*)(Fp + k0);        // A 16x4 frag: K = k0 + half*2 +{0,1}
    v2f b0 = *(const v2f*)(Wp + 0);        // B 4x16 frag, n-subtile 0
    v2f b1 = *(const v2f*)(Wp + 32);       // n-subtile 1 (+16 cols)
    v2f b2 = *(const v2f*)(Wp + 64);       // n-subtile 2 (+32 cols)
    Wp += (size_t)2 * NCOLP * 2;           // next k-quad (2 swizzled rows)
    c0 = __builtin_amdgcn_wmma_f32_16x16x4_f32(false, a, false, b0, (short)0,
                                               c0, false, false);
    c1 = __builtin_amdgcn_wmma_f32_16x16x4_f32(false, a, false, b1, (short)0,
                                               c1, false, false);
    c2 = __builtin_amdgcn_wmma_f32_16x16x4_f32(false, a, false, b2, (short)0,
                                               c2, false, false);
  }

  // C/D layout: VGPR r -> M = r + half*8; lane&15 -> N. Stage tile to LDS.
#pragma unroll
  for (int r = 0; r < 8; ++r) {
    int m = r + half * 8;
    tileL[m * 48 + 0 + col] = c0[r];
    tileL[m * 48 + 16 + col] = c1[r];
    tileL[m * 48 + 32 + col] = c2[r];
  }
  __asm__ volatile("s_wait_dscnt 0" ::: "memory");

  // LBS epilogue: 8 iters x 32 lanes cover 16 batches x 16 vertices
  for (int i = 0; i < 8; ++i) {
    int ml = 2 * i + half;       // local batch row
    int vl = col;                // local vertex
    int v = v0 + vl;
    if (v < V_) {
      int b = m0 + ml;
      float x = tileL[ml * 48 + vl * 3 + 0] + vtemp[v * 3 + 0];
      float y = tileL[ml * 48 + vl * 3 + 1] + vtemp[v * 3 + 1];
      float z = tileL[ml * 48 + vl * 3 + 2] + vtemp[v * 3 + 2];
      const float* wv = lbsw + v * 5;
      float T[12];
#pragma unroll
      for (int r = 0; r < 12; ++r) T[r] = 0.f;
#pragma unroll
      for (int j = 0; j < 5; ++j) {
        float wj = wv[j];
        const float* Ab = aL + ml * 60 + j * 12;
#pragma unroll
        for (int r = 0; r < 12; ++r) T[r] += wj * Ab[r];
      }
      float ox = T[0] * x + T[1] * y + T[2] * z + T[3];
      float oy = T[4] * x + T[5] * y + T[6] * z + T[7];
      float oz = T[8] * x + T[9] * y + T[10] * z + T[11];
      float* o = verts + ((size_t)b * V_ + v) * 3;
      o[0] = ox; o[1] = oy; o[2] = oz;
    }
  }
}

// ---------------------------------------------------------------------------
extern "C" void kernel_launch(void* const* d_in, const int* in_sizes, int n_in,
                              void* d_out, int out_size, void* d_ws,
                              size_t ws_size, hipStream_t stream) {
  const float* expr = (const float*)d_in[0];
  const float* pose = (const float*)d_in[1];
  const float* vtemp = (const float*)d_in[2];
  const float* shapedirs = (const float*)d_in[3];
  const float* posedirs = (const float*)d_in[4];
  const float* Jreg = (const float*)d_in[5];
  const float* lbsw = (const float*)d_in[6];

  float* out = (float*)d_out;
  float* verts = out;             // [2048,5023,3]
  float* pf = out + PF_OFF;       // [2048,36]
  float* Aout = out + A_OFF;      // [2048,5,4,4]

  float* ws = (float*)d_ws;
  float* F = ws + WS_F;
  float* Wsw = ws + WS_W;
  float* jt = ws + WS_JT;
  float* js = ws + WS_JS;

  build_joint_bases<<<3, 256, 0, stream>>>(Jreg, vtemp, shapedirs, jt, js);
  build_W<<<(KF * NCOLP + 255) / 256, 256, 0, stream>>>(shapedirs, posedirs,
                                                        Wsw);
  flame_pose_fk<<<B_ / 256, 256, 0, stream>>>(expr, pose, jt, js, F, pf, Aout);
  flame_gemm_lbs<<<(MT * NT) / 8, 256, 0, stream>>>(F, Wsw, Aout, vtemp, lbsw,
                                                    verts);
}